// MixtureOfExperts_71494025609399
// MI455X (gfx1250) — compile-verified
//
#include <hip/hip_runtime.h>

// MoE grouped-GEMM for MI455X (gfx1250, wave32), fp32 WMMA path.
// out[n] = sum_k p[n,k] * (x[n] . W[idx[n,k]] + b[idx[n,k]])
//
// N=16384 tokens, D=512, E=8 experts, K=2 top-k. Compute-bound in fp32
// (17.2 GFLOP vs ~72MB traffic); x and W are L2-resident (192MB L2).

#define N_TOK   16384
#define DIM     512
#define NEXP    8
#define TOPK    2
#define NK      (N_TOK * TOPK)

#define MT      64          // tokens per block tile
#define OT      128         // outputs per block (8 waves * 16 cols)
#define XSTRIDE 516         // LDS row stride in dwords (pad: banks 4l, conflict-free)

typedef __attribute__((ext_vector_type(2))) float v2f;
typedef __attribute__((ext_vector_type(8))) float v8f;

// ---------------------------------------------------------------- zero output
__global__ void moe_zero(float4* __restrict__ out) {
    int i = blockIdx.x * blockDim.x + threadIdx.x;
    float4 z = {0.0f, 0.0f, 0.0f, 0.0f};
    out[i] = z;
}

// ------------------------------------------------- deterministic expert routing
// One wave32 per expert scans all (token,slot) entries in order and compacts
// the matches with ballot + prefix-popcount. lists[e][j] = n*TOPK + k.
__global__ void moe_route(const int* __restrict__ idx,
                          int* __restrict__ lists,
                          int* __restrict__ counts) {
    const int e    = blockIdx.x;
    const int lane = threadIdx.x;        // 0..31
    int base = 0;
    for (int i0 = 0; i0 < NK; i0 += 32) {        // NK is a multiple of 32
        const int i = i0 + lane;
        const bool m = (idx[i] == e);
        const unsigned mask = (unsigned)__ballot(m);   // wave32: low 32 bits
        const int pos = __popc(mask & ((1u << lane) - 1u));
        if (m) lists[e * N_TOK + base + pos] = i;
        base += __popc(mask);
    }
    if (lane == 0) counts[e] = base;
}

// ------------------------------------------------------------- grouped GEMM
// Block: 256 threads (8 waves). Tile: 64 gathered tokens x 128 outputs.
// Wave: 16-column output strip, 4 stacked 16x16 accumulators (M-tiles).
// Inner loop per K-step-of-4: 4x ds_load_b64 (A) + 2x global_load_b32 (B)
// + 4x v_wmma_f32_16x16x4_f32.
__global__ __launch_bounds__(256)
void moe_gemm(const float* __restrict__ x,
              const float* __restrict__ prob,
              const float* __restrict__ W,
              const float* __restrict__ bias,
              const int*   __restrict__ lists,
              const int*   __restrict__ counts,
              float*       __restrict__ out) {
    __shared__ float xs[MT * XSTRIDE];

    const int e   = blockIdx.z;
    const int cnt = counts[e];
    const int m0  = blockIdx.x * MT;
    if (m0 >= cnt) return;

    const int* list = lists + e * N_TOK + m0;
    const int  rows = min(MT, cnt - m0);

    // ---- gather 64 token rows of x into LDS (zero-fill past 'rows') ----
    const int tid = threadIdx.x;
    for (int i = tid; i < MT * (DIM / 4); i += 256) {
        const int r  = i >> 7;          // DIM/4 == 128 float4 per row
        const int c4 = i & 127;
        float4 v = {0.0f, 0.0f, 0.0f, 0.0f};
        if (r < rows) {
            const int n = list[r] >> 1;                  // entry = n*2 + k
            v = ((const float4*)(x + (size_t)n * DIM))[c4];
        }
        *(float4*)&xs[r * XSTRIDE + c4 * 4] = v;
    }
    __syncthreads();

    const int lane = tid & 31;
    const int wave = tid >> 5;
    const int lcol = lane & 15;
    const int koff = (lane >> 4) << 1;                   // 0 (lanes 0-15) or 2
    const int o    = blockIdx.y * OT + wave * 16 + lcol; // this lane's column

    const float* wp = W + (size_t)e * DIM * DIM + o;     // W[e][k][o], stride DIM

    v8f acc[4] = {};

    for (int kk = 0; kk < DIM; kk += 4) {
        const int kr = kk + koff;
        // B fragment: rows kr, kr+1 of W[e], column o
        v2f b;
        b.x = wp[(size_t)kr * DIM];
        b.y = wp[(size_t)(kr + 1) * DIM];
        // A fragments: 4 M-tiles, row = mtile*16 + lcol, K = kr..kr+1
        const int abase = lcol * XSTRIDE + kr;
        v2f a0 = *(const v2f*)&xs[abase + 0  * 16 * XSTRIDE];
        v2f a1 = *(const v2f*)&xs[abase + 1  * 16 * XSTRIDE];
        v2f a2 = *(const v2f*)&xs[abase + 2  * 16 * XSTRIDE];
        v2f a3 = *(const v2f*)&xs[abase + 3  * 16 * XSTRIDE];
        acc[0] = __builtin_amdgcn_wmma_f32_16x16x4_f32(false, a0, false, b,
                                                       (short)0, acc[0], false, false);
        acc[1] = __builtin_amdgcn_wmma_f32_16x16x4_f32(false, a1, false, b,
                                                       (short)0, acc[1], false, false);
        acc[2] = __builtin_amdgcn_wmma_f32_16x16x4_f32(false, a2, false, b,
                                                       (short)0, acc[2], false, false);
        acc[3] = __builtin_amdgcn_wmma_f32_16x16x4_f32(false, a3, false, b,
                                                       (short)0, acc[3], false, false);
    }

    // ---- epilogue: out[n][o] += p * (acc + bias[e][o]) -------------------
    // C/D layout: VGPR r -> M = r + (lane>=16 ? 8 : 0), N = lane&15.
    const float bo    = bias[e * DIM + o];
    const int   rbase = (lane >> 4) * 8;
#pragma unroll
    for (int t = 0; t < 4; ++t) {
#pragma unroll
        for (int r = 0; r < 8; ++r) {
            const int row = t * 16 + rbase + r;
            if (row < rows) {
                const int entry = list[row];
                const int n = entry >> 1;
                const float p = prob[entry];             // prob[n*TOPK + k]
                atomicAdd(&out[(size_t)n * DIM + o], p * (acc[t][r] + bo));
            }
        }
    }
}

// --------------------------------------------------------------------- launch
extern "C" void kernel_launch(void* const* d_in, const int* in_sizes, int n_in,
                              void* d_out, int out_size, void* d_ws, size_t ws_size,
                              hipStream_t stream) {
    const float* x    = (const float*)d_in[0];   // [N, D]
    const float* prob = (const float*)d_in[1];   // [N, K]
    const int*   idx  = (const int*)  d_in[2];   // [N, K]
    const float* W    = (const float*)d_in[3];   // [E, D, D]
    const float* bias = (const float*)d_in[4];   // [E, D]
    float*       out  = (float*)d_out;           // [N, D]

    // ws layout: lists[E*N] ints, then counts[E] ints (~512 KB)
    int* lists  = (int*)d_ws;
    int* counts = lists + NEXP * N_TOK;

    moe_zero<<<(N_TOK * DIM / 4) / 256, 256, 0, stream>>>((float4*)out);
    moe_route<<<NEXP, 32, 0, stream>>>(idx, lists, counts);

    dim3 grid(N_TOK / MT, DIM / OT, NEXP);       // (256, 4, 8); empty tiles exit
    moe_gemm<<<grid, 256, 0, stream>>>(x, prob, W, bias, lists, counts, out);
}